// GAT_Net_52261162057815
// MI455X (gfx1250) — compile-verified
//
#include <hip/hip_runtime.h>
#include <hip/hip_bf16.h>
#include <math.h>

// ---------------------------------------------------------------------------
// 2-layer GAT for MI455X (gfx1250, wave32).
//  - GEMMs via V_WMMA_F32_16X16X4_F32 (fp32 A/B, fp32 acc -> exact numerics)
//  - LDS weight tiles stored K-pair-interleaved so each B fragment is one
//    aligned ds_load_b64 straight into the WMMA operand pair (no v_mov fixup)
//  - segment softmax via 3 edge passes with L2-resident atomics
// ---------------------------------------------------------------------------

typedef __attribute__((ext_vector_type(2))) float v2f;
typedef __attribute__((ext_vector_type(8))) float v8f;

#define F_IN  512
#define C1    64      // HEADS*HID
#define HEADS 8
#define HID   8
#define CLS   40
#define NEG   0.2f

// -------- order-preserving float <-> uint encoding for atomic max ----------
__device__ __forceinline__ unsigned fenc(float f) {
  unsigned b = __float_as_uint(f);
  return (b & 0x80000000u) ? ~b : (b | 0x80000000u);
}
__device__ __forceinline__ float fdec(unsigned u) {
  unsigned b = (u & 0x80000000u) ? (u & 0x7FFFFFFFu) : ~u;
  return __uint_as_float(b);
}
#define ENC_NEG_INF 0x007FFFFFu   // fenc(-inf)

__device__ __forceinline__ float leaky(float v) { return v > 0.f ? v : NEG * v; }

// ---------------------------------------------------------------------------
// GEMM1: h[N,64] = x[N,512] @ W1[512,64]   (wave: 16 rows x 64 cols)
// LDS layout: pair p = k>>1 ; sW[p*128 + n*2 + (k&1)] = W1chunk[k][n]
// ---------------------------------------------------------------------------
__global__ __launch_bounds__(256) void k_gemm1(const float* __restrict__ x,
                                               const float* __restrict__ W1,
                                               float* __restrict__ h, int N) {
  __shared__ float sW[128 * 64];                 // 32 KB K-chunk, pair-swizzled
  const int tid  = threadIdx.x;
  const int wv   = tid >> 5;
  const int lane = tid & 31;
  const int row0 = blockIdx.x * 128 + wv * 16;
  const int mi   = lane & 15;
  const int kb   = (lane >> 4) * 2;              // K-pair base: 0 or 2
  const int kp   = lane >> 4;                    // which pair within K4 group
  int arow = row0 + mi;
  if (arow >= N) arow = N - 1;                   // clamp; stores are guarded
  const float* xrow = x + (size_t)arow * F_IN;

  v8f a0 = {}, a1 = {}, a2 = {}, a3 = {};
  for (int kc = 0; kc < F_IN; kc += 128) {
    __syncthreads();
    // swizzle-stage: 64 K-pairs x 16 col-quads, interleave row k / k+1
    for (int it = tid; it < 64 * 16; it += 256) {
      const int p  = it >> 4;
      const int n4 = (it & 15) * 4;
      const float* g0 = W1 + (size_t)(kc + p * 2) * C1 + n4;
      float4 r0 = *(const float4*)g0;
      float4 r1 = *(const float4*)(g0 + C1);
      float* sp = sW + p * 128 + n4 * 2;
      *(float2*)(sp + 0) = make_float2(r0.x, r1.x);
      *(float2*)(sp + 2) = make_float2(r0.y, r1.y);
      *(float2*)(sp + 4) = make_float2(r0.z, r1.z);
      *(float2*)(sp + 6) = make_float2(r0.w, r1.w);
    }
    __syncthreads();
#pragma unroll 4
    for (int k = 0; k < 128; k += 4) {
      v2f a = *(const v2f*)(xrow + kc + k + kb);       // rows mi, K kb..kb+1
      const float* bp = sW + ((k >> 1) + kp) * 128 + mi * 2;
      v2f b0 = *(const v2f*)(bp);                      // cols mi
      v2f b1 = *(const v2f*)(bp + 32);                 // cols 16+mi
      v2f b2 = *(const v2f*)(bp + 64);                 // cols 32+mi
      v2f b3 = *(const v2f*)(bp + 96);                 // cols 48+mi
      a0 = __builtin_amdgcn_wmma_f32_16x16x4_f32(false, a, false, b0, (short)0, a0, false, false);
      a1 = __builtin_amdgcn_wmma_f32_16x16x4_f32(false, a, false, b1, (short)0, a1, false, false);
      a2 = __builtin_amdgcn_wmma_f32_16x16x4_f32(false, a, false, b2, (short)0, a2, false, false);
      a3 = __builtin_amdgcn_wmma_f32_16x16x4_f32(false, a, false, b3, (short)0, a3, false, false);
    }
  }
  const int mo = (lane >> 4) * 8;                // C/D: vgpr r -> row r(+8)
#pragma unroll
  for (int r = 0; r < 8; ++r) {
    int row = row0 + mo + r;
    if (row < N) {
      float* hp = h + (size_t)row * C1 + mi;
      hp[0] = a0[r]; hp[16] = a1[r]; hp[32] = a2[r]; hp[48] = a3[r];
    }
  }
}

// ---------------------------------------------------------------------------
// GEMM2: g[N,40] = helu[N,64] @ W2[64,40]  (3 N-tiles, last guarded)
// LDS layout: pair p = k>>1 ; sW[p*80 + n*2 + (k&1)] = W2[k][n]
// ---------------------------------------------------------------------------
__global__ __launch_bounds__(256) void k_gemm2(const float* __restrict__ h,
                                               const float* __restrict__ W2,
                                               float* __restrict__ g, int N) {
  __shared__ float sW[32 * 80];                  // 10 KB, pair-swizzled
  const int tid = threadIdx.x;
  for (int it = tid; it < 32 * 10; it += 256) {  // 32 K-pairs x 10 col-quads
    const int p  = it / 10;
    const int n4 = (it % 10) * 4;
    const float* g0 = W2 + (size_t)(p * 2) * CLS + n4;
    float4 r0 = *(const float4*)g0;
    float4 r1 = *(const float4*)(g0 + CLS);
    float* sp = sW + p * 80 + n4 * 2;
    *(float2*)(sp + 0) = make_float2(r0.x, r1.x);
    *(float2*)(sp + 2) = make_float2(r0.y, r1.y);
    *(float2*)(sp + 4) = make_float2(r0.z, r1.z);
    *(float2*)(sp + 6) = make_float2(r0.w, r1.w);
  }
  __syncthreads();

  const int wv = tid >> 5, lane = tid & 31;
  const int row0 = blockIdx.x * 128 + wv * 16;
  const int mi = lane & 15, kb = (lane >> 4) * 2, kp = lane >> 4;
  int arow = row0 + mi;
  if (arow >= N) arow = N - 1;
  const float* hrow = h + (size_t)arow * C1;
  const bool c2ok = (32 + mi) < CLS;
  const v2f zz = {};

  v8f a0 = {}, a1 = {}, a2 = {};
#pragma unroll
  for (int k = 0; k < 64; k += 4) {
    v2f a = *(const v2f*)(hrow + k + kb);
    const float* bp = sW + ((k >> 1) + kp) * 80 + mi * 2;
    v2f b0 = *(const v2f*)(bp);
    v2f b1 = *(const v2f*)(bp + 32);
    v2f b2 = c2ok ? *(const v2f*)(bp + 64) : zz;
    a0 = __builtin_amdgcn_wmma_f32_16x16x4_f32(false, a, false, b0, (short)0, a0, false, false);
    a1 = __builtin_amdgcn_wmma_f32_16x16x4_f32(false, a, false, b1, (short)0, a1, false, false);
    a2 = __builtin_amdgcn_wmma_f32_16x16x4_f32(false, a, false, b2, (short)0, a2, false, false);
  }
  const int mo = (lane >> 4) * 8;
#pragma unroll
  for (int r = 0; r < 8; ++r) {
    int row = row0 + mo + r;
    if (row >= N) continue;
    float* gp = g + (size_t)row * CLS;
    gp[mi]      = a0[r];
    gp[16 + mi] = a1[r];
    if (c2ok) gp[32 + mi] = a2[r];
  }
}

// ---------------------------------------------------------------------------
// per-node attention halves
// ---------------------------------------------------------------------------
__global__ void k_attn1(const float* __restrict__ h, const float* __restrict__ as1,
                        const float* __restrict__ ad1, float* __restrict__ al,
                        float* __restrict__ ar, int N) {
  int i = blockIdx.x * blockDim.x + threadIdx.x;     // (node, head)
  if (i >= N * HEADS) return;
  int n = i >> 3, hd = i & 7;
  const float* hp = h + (size_t)n * C1 + hd * HID;
  const float* sp = as1 + hd * HID;
  const float* dp = ad1 + hd * HID;
  float sl = 0.f, sr = 0.f;
#pragma unroll
  for (int c = 0; c < HID; ++c) { sl += hp[c] * sp[c]; sr += hp[c] * dp[c]; }
  al[i] = sl; ar[i] = sr;
}

__global__ void k_attn2(const float* __restrict__ g, const float* __restrict__ as2,
                        const float* __restrict__ ad2, float* __restrict__ al,
                        float* __restrict__ ar, int N) {
  int n = blockIdx.x * blockDim.x + threadIdx.x;
  if (n >= N) return;
  const float* gp = g + (size_t)n * CLS;
  float sl = 0.f, sr = 0.f;
#pragma unroll 8
  for (int c = 0; c < CLS; ++c) { sl += gp[c] * as2[c]; sr += gp[c] * ad2[c]; }
  al[n] = sl; ar[n] = sr;
}

// ---------------------------------------------------------------------------
// edge passes, layer 1 (8 heads).  threads cover E real edges + N self loops
// ---------------------------------------------------------------------------
__device__ __forceinline__ void edge_sd(const int* src, const int* dst, int E,
                                        int i, int& s, int& d) {
  if (i < E) { s = src[i]; d = dst[i]; } else { s = i - E; d = s; }
}

__global__ void k_emax1(const int* __restrict__ src, const int* __restrict__ dst,
                        const float* __restrict__ al, const float* __restrict__ ar,
                        unsigned* __restrict__ m, int E, int N) {
  int i = blockIdx.x * blockDim.x + threadIdx.x;
  if (i >= E + N) return;
  int s, d; edge_sd(src, dst, E, i, s, d);
#pragma unroll
  for (int hd = 0; hd < HEADS; ++hd) {
    float v = leaky(al[s * HEADS + hd] + ar[d * HEADS + hd]);
    atomicMax(&m[d * HEADS + hd], fenc(v));
  }
}

__global__ void k_esum1(const int* __restrict__ src, const int* __restrict__ dst,
                        const float* __restrict__ al, const float* __restrict__ ar,
                        const unsigned* __restrict__ m, float* __restrict__ z,
                        int E, int N) {
  int i = blockIdx.x * blockDim.x + threadIdx.x;
  if (i >= E + N) return;
  int s, d; edge_sd(src, dst, E, i, s, d);
#pragma unroll
  for (int hd = 0; hd < HEADS; ++hd) {
    float v = leaky(al[s * HEADS + hd] + ar[d * HEADS + hd]);
    atomicAdd(&z[d * HEADS + hd], expf(v - fdec(m[d * HEADS + hd])));
  }
}

__global__ void k_eagg1(const int* __restrict__ src, const int* __restrict__ dst,
                        const float* __restrict__ al, const float* __restrict__ ar,
                        const unsigned* __restrict__ m, const float* __restrict__ z,
                        const float* __restrict__ h, float* __restrict__ agg,
                        int E, int N) {
  int i = blockIdx.x * blockDim.x + threadIdx.x;
  if (i >= E + N) return;
  int s, d; edge_sd(src, dst, E, i, s, d);
#pragma unroll
  for (int hd = 0; hd < HEADS; ++hd) {
    float v = leaky(al[s * HEADS + hd] + ar[d * HEADS + hd]);
    float alpha = expf(v - fdec(m[d * HEADS + hd])) / z[d * HEADS + hd];
    const float4* hs = (const float4*)(h + (size_t)s * C1 + hd * HID);
    float4 p0 = hs[0], p1 = hs[1];
    float* op = agg + (size_t)d * C1 + hd * HID;
    atomicAdd(op + 0, p0.x * alpha); atomicAdd(op + 1, p0.y * alpha);
    atomicAdd(op + 2, p0.z * alpha); atomicAdd(op + 3, p0.w * alpha);
    atomicAdd(op + 4, p1.x * alpha); atomicAdd(op + 5, p1.y * alpha);
    atomicAdd(op + 6, p1.z * alpha); atomicAdd(op + 7, p1.w * alpha);
  }
}

// bias + ELU  (writes layer-2 input)
__global__ void k_fin1(const float* __restrict__ agg, const float* __restrict__ b1,
                       float* __restrict__ helu, int N) {
  int i = blockIdx.x * blockDim.x + threadIdx.x;
  if (i >= N * C1) return;
  float v = agg[i] + b1[i & (C1 - 1)];
  helu[i] = v > 0.f ? v : expf(v) - 1.f;
}

// ---------------------------------------------------------------------------
// edge passes, layer 2 (1 head, 40 ch)
// ---------------------------------------------------------------------------
__global__ void k_emax2(const int* __restrict__ src, const int* __restrict__ dst,
                        const float* __restrict__ al, const float* __restrict__ ar,
                        unsigned* __restrict__ m, int E, int N) {
  int i = blockIdx.x * blockDim.x + threadIdx.x;
  if (i >= E + N) return;
  int s, d; edge_sd(src, dst, E, i, s, d);
  atomicMax(&m[d], fenc(leaky(al[s] + ar[d])));
}

__global__ void k_esum2(const int* __restrict__ src, const int* __restrict__ dst,
                        const float* __restrict__ al, const float* __restrict__ ar,
                        const unsigned* __restrict__ m, float* __restrict__ z,
                        int E, int N) {
  int i = blockIdx.x * blockDim.x + threadIdx.x;
  if (i >= E + N) return;
  int s, d; edge_sd(src, dst, E, i, s, d);
  atomicAdd(&z[d], expf(leaky(al[s] + ar[d]) - fdec(m[d])));
}

__global__ void k_eagg2(const int* __restrict__ src, const int* __restrict__ dst,
                        const float* __restrict__ al, const float* __restrict__ ar,
                        const unsigned* __restrict__ m, const float* __restrict__ z,
                        const float* __restrict__ g, float* __restrict__ out,
                        int E, int N) {
  int i = blockIdx.x * blockDim.x + threadIdx.x;
  if (i >= E + N) return;
  int s, d; edge_sd(src, dst, E, i, s, d);
  float alpha = expf(leaky(al[s] + ar[d]) - fdec(m[d])) / z[d];
  const float4* gs = (const float4*)(g + (size_t)s * CLS);
  float* op = out + (size_t)d * CLS;
#pragma unroll
  for (int q = 0; q < CLS / 4; ++q) {
    float4 p = gs[q];
    atomicAdd(op + 4 * q + 0, p.x * alpha);
    atomicAdd(op + 4 * q + 1, p.y * alpha);
    atomicAdd(op + 4 * q + 2, p.z * alpha);
    atomicAdd(op + 4 * q + 3, p.w * alpha);
  }
}

// bias + log_softmax, in place on d_out
__global__ void k_fin2(float* __restrict__ out, const float* __restrict__ b2, int N) {
  int n = blockIdx.x * blockDim.x + threadIdx.x;
  if (n >= N) return;
  float* op = out + (size_t)n * CLS;
  float v[CLS];
  float mx = -3.4e38f;
#pragma unroll 8
  for (int c = 0; c < CLS; ++c) { v[c] = op[c] + b2[c]; mx = fmaxf(mx, v[c]); }
  float se = 0.f;
#pragma unroll 8
  for (int c = 0; c < CLS; ++c) se += expf(v[c] - mx);
  float ls = mx + logf(se);
#pragma unroll 8
  for (int c = 0; c < CLS; ++c) op[c] = v[c] - ls;
}

__global__ void k_fill_u32(unsigned* __restrict__ p, unsigned val, int n) {
  int i = blockIdx.x * blockDim.x + threadIdx.x;
  if (i < n) p[i] = val;
}

// ---------------------------------------------------------------------------
extern "C" void kernel_launch(void* const* d_in, const int* in_sizes, int n_in,
                              void* d_out, int out_size, void* d_ws, size_t ws_size,
                              hipStream_t stream) {
  const float* x    = (const float*)d_in[0];
  const int*   esrc = (const int*)  d_in[1];
  const int*   edst = (const int*)  d_in[2];
  const float* W1   = (const float*)d_in[3];
  const float* as1  = (const float*)d_in[4];
  const float* ad1  = (const float*)d_in[5];
  const float* b1   = (const float*)d_in[6];
  const float* W2   = (const float*)d_in[7];
  const float* as2  = (const float*)d_in[8];
  const float* ad2  = (const float*)d_in[9];
  const float* b2   = (const float*)d_in[10];

  const int N  = in_sizes[0] / F_IN;
  const int E  = in_sizes[1];
  const int ET = E + N;                 // with self loops
  float* out = (float*)d_out;

  // workspace carve-out
  size_t off = 0;
  auto carve = [&](size_t nfloats) {
    void* p = (char*)d_ws + off;
    off += nfloats * sizeof(float);
    return p;
  };
  float*    h_raw = (float*)   carve((size_t)N * C1);   // x@W1, reused for ELU(h)
  float*    agg1  = (float*)   carve((size_t)N * C1);
  float*    al1   = (float*)   carve((size_t)N * HEADS);
  float*    ar1   = (float*)   carve((size_t)N * HEADS);
  float*    z1    = (float*)   carve((size_t)N * HEADS);
  unsigned* m1    = (unsigned*)carve((size_t)N * HEADS);
  float*    gbuf  = (float*)   carve((size_t)N * CLS);
  float*    al2   = (float*)   carve((size_t)N);
  float*    ar2   = (float*)   carve((size_t)N);
  float*    z2    = (float*)   carve((size_t)N);
  unsigned* m2    = (unsigned*)carve((size_t)N);

  const int TB = 256;
  const int gN128 = (N + 127) / 128;
  const int gE    = (ET + TB - 1) / TB;

  // init accumulators (graph-capture-safe, stream-ordered)
  hipMemsetAsync(agg1, 0, (size_t)N * C1 * sizeof(float), stream);
  hipMemsetAsync(z1,   0, (size_t)N * HEADS * sizeof(float), stream);
  hipMemsetAsync(z2,   0, (size_t)N * sizeof(float), stream);
  hipMemsetAsync(out,  0, (size_t)N * CLS * sizeof(float), stream);
  k_fill_u32<<<(N * HEADS + TB - 1) / TB, TB, 0, stream>>>(m1, ENC_NEG_INF, N * HEADS);
  k_fill_u32<<<(N + TB - 1) / TB, TB, 0, stream>>>(m2, ENC_NEG_INF, N);

  // ---- layer 1 ----
  k_gemm1<<<gN128, TB, 0, stream>>>(x, W1, h_raw, N);
  k_attn1<<<(N * HEADS + TB - 1) / TB, TB, 0, stream>>>(h_raw, as1, ad1, al1, ar1, N);
  k_emax1<<<gE, TB, 0, stream>>>(esrc, edst, al1, ar1, m1, E, N);
  k_esum1<<<gE, TB, 0, stream>>>(esrc, edst, al1, ar1, m1, z1, E, N);
  k_eagg1<<<gE, TB, 0, stream>>>(esrc, edst, al1, ar1, m1, z1, h_raw, agg1, E, N);
  k_fin1<<<((size_t)N * C1 + TB - 1) / TB, TB, 0, stream>>>(agg1, b1, h_raw, N);

  // ---- layer 2 ----
  k_gemm2<<<gN128, TB, 0, stream>>>(h_raw, W2, gbuf, N);
  k_attn2<<<(N + TB - 1) / TB, TB, 0, stream>>>(gbuf, as2, ad2, al2, ar2, N);
  k_emax2<<<gE, TB, 0, stream>>>(esrc, edst, al2, ar2, m2, E, N);
  k_esum2<<<gE, TB, 0, stream>>>(esrc, edst, al2, ar2, m2, z2, E, N);
  k_eagg2<<<gE, TB, 0, stream>>>(esrc, edst, al2, ar2, m2, z2, gbuf, out, E, N);
  k_fin2<<<(N + TB - 1) / TB, TB, 0, stream>>>(out, b2, N);
}